// NoisyTopkRouter_31937376813282
// MI455X (gfx1250) — compile-verified
//
#include <hip/hip_runtime.h>
#include <math.h>

// ---------------------------------------------------------------------------
// NoisyTopkRouter for MI455X (gfx1250, wave32).
// Memory-bound: 134 MB fp32 activations streamed once (23.3 TB/s -> ~6us).
// GEMM (N=16 experts) via V_WMMA_F32_16X16X4_F32. Split-K x4 + b128 loads +
// explicit 8-deep software pipeline => ~16 b128 loads in flight per wave,
// ~32 MB in flight across 4096 waves (covers BW*latency ~= 16 MB).
// ---------------------------------------------------------------------------

typedef __attribute__((ext_vector_type(2))) float v2f;
typedef __attribute__((ext_vector_type(4))) float v4f;
typedef __attribute__((ext_vector_type(8))) float v8f;

#define D_DIM   2048
#define E_DIM   16
#define NTOK    16384                 // B*T = 4*4096
#define NTILE   (NTOK / 16)           // 1024 tiles of 16 tokens
#define SPLITK  4                     // waves per tile (K split)
#define KCHUNK  (D_DIM / SPLITK)      // 512 floats of K per wave
#define TPB     (SPLITK * 32)         // 128 threads/block, 1 tile/block
#define PDEPTH  8                     // software pipeline stages (a4/b4 pairs)
#define STEP    8                     // K floats consumed per stage

#define ROUT_CNT  (NTOK * E_DIM)      // 262144 floats
#define IDX_CNT   (NTOK * 2)          // 32768 floats (indices stored as float)

__device__ __forceinline__ float fnoise(float x) {
    // f(x) = sign(x) * sqrt(|x|); sign(0)=0 is preserved since sqrt(0)=0.
    return copysignf(sqrtf(fabsf(x)), x);
}

// ---------------------------------------------------------------------------
// Kernel 1: w_noisy[e][d] = W + sigma_W * f(eps_out[e]) * f(eps_in[d]);
//           b_noisy[e]    = b + sigma_b * f(eps_out[e])
// ---------------------------------------------------------------------------
__global__ __launch_bounds__(256) void prep_kernel(
    const float* __restrict__ weight, const float* __restrict__ sigma_weight,
    const float* __restrict__ bias,   const float* __restrict__ sigma_bias,
    const float* __restrict__ eps_in, const float* __restrict__ eps_out,
    float* __restrict__ w_noisy, float* __restrict__ b_noisy) {
    int idx = blockIdx.x * blockDim.x + threadIdx.x;
    if (idx < E_DIM * D_DIM) {
        int e = idx >> 11;         // / 2048
        int d = idx & (D_DIM - 1);
        float fo = fnoise(eps_out[e]);
        float fi = fnoise(eps_in[d]);
        w_noisy[idx] = weight[idx] + sigma_weight[idx] * fo * fi;
    }
    if (idx < E_DIM) {
        b_noisy[idx] = bias[idx] + sigma_bias[idx] * fnoise(eps_out[idx]);
    }
}

// ---------------------------------------------------------------------------
// Kernel 2: one 16-token tile per block; 4 waves split the K=2048 reduction.
//
// f32 WMMA layouts (ISA 7.12.2):
//   A 16x4 : lane l -> row M=l%16, K-slot = 2*(l/16) + vgpr
//   B 4x16 : lane l -> col N=l%16, K-slot = 2*(l/16) + vgpr
//   C 16x16: vgpr r, lanes 0-15 -> M=r ; lanes 16-31 -> M=r+8 ; N = lane%16
//
// b128 trick: lane l loads float4 at physical k0 + 4*(l>>4) for BOTH A and B.
// WMMA#0 eats .xy, WMMA#1 eats .zw. The physical-K -> K-slot mapping is a
// permutation applied identically to A and B, so the dot product is exact.
// ---------------------------------------------------------------------------
__global__ __launch_bounds__(TPB) void router_kernel(
    const float* __restrict__ mh,       // [NTOK, D]
    const float* __restrict__ w_noisy,  // [E, D]
    const float* __restrict__ b_noisy,  // [E]
    float* __restrict__ rout,           // [NTOK, E]
    float* __restrict__ idx_out,        // [NTOK, 2] (as float)
    float* __restrict__ partials)       // [NTILE] sum of lse^2 per tile
{
    __shared__ v8f lds_part[(SPLITK - 1) * 32];

    const int lane = threadIdx.x & 31;
    const int wid  = threadIdx.x >> 5;       // K-split wave id 0..3
    const int tile = blockIdx.x;
    const int e    = lane & 15;              // expert col for B/C; token row for A
    const int half = lane >> 4;              // 0 or 1
    const int koff = wid * KCHUNK + half * 4;

    const float* arow = mh      + (size_t)(tile * 16 + e) * D_DIM + koff;
    const float* brow = w_noisy + (size_t)e * D_DIM + koff;

    // ---- explicit software pipeline: PDEPTH a4/b4 pairs in flight ----
    v4f abuf[PDEPTH], bbuf[PDEPTH];
    #pragma unroll
    for (int i = 0; i < PDEPTH; ++i) {
        abuf[i] = *(const v4f*)(arow + i * STEP);
        bbuf[i] = *(const v4f*)(brow + i * STEP);
    }

    v8f acc = {};
    for (int k0 = PDEPTH * STEP; k0 < KCHUNK; k0 += PDEPTH * STEP) {
        #pragma unroll
        for (int i = 0; i < PDEPTH; ++i) {
            v4f a4 = abuf[i];
            v4f b4 = bbuf[i];
            abuf[i] = *(const v4f*)(arow + k0 + i * STEP);
            bbuf[i] = *(const v4f*)(brow + k0 + i * STEP);
            v2f alo = __builtin_shufflevector(a4, a4, 0, 1);
            v2f blo = __builtin_shufflevector(b4, b4, 0, 1);
            v2f ahi = __builtin_shufflevector(a4, a4, 2, 3);
            v2f bhi = __builtin_shufflevector(b4, b4, 2, 3);
            acc = __builtin_amdgcn_wmma_f32_16x16x4_f32(false, alo, false, blo, (short)0, acc, false, false);
            acc = __builtin_amdgcn_wmma_f32_16x16x4_f32(false, ahi, false, bhi, (short)0, acc, false, false);
        }
    }
    // ---- drain the pipeline ----
    #pragma unroll
    for (int i = 0; i < PDEPTH; ++i) {
        v4f a4 = abuf[i];
        v4f b4 = bbuf[i];
        v2f alo = __builtin_shufflevector(a4, a4, 0, 1);
        v2f blo = __builtin_shufflevector(b4, b4, 0, 1);
        v2f ahi = __builtin_shufflevector(a4, a4, 2, 3);
        v2f bhi = __builtin_shufflevector(b4, b4, 2, 3);
        acc = __builtin_amdgcn_wmma_f32_16x16x4_f32(false, alo, false, blo, (short)0, acc, false, false);
        acc = __builtin_amdgcn_wmma_f32_16x16x4_f32(false, ahi, false, bhi, (short)0, acc, false, false);
    }

    // -------- combine split-K partials through LDS (deterministic) --------
    if (wid != 0) lds_part[(wid - 1) * 32 + lane] = acc;
    __syncthreads();
    if (wid != 0) return;

    #pragma unroll
    for (int w = 0; w < SPLITK - 1; ++w) {
        v8f p = lds_part[w * 32 + lane];
        #pragma unroll
        for (int r = 0; r < 8; ++r) acc[r] += p[r];
    }

    const float bn = b_noisy[e];
    float lse2_sum = 0.0f;

    #pragma unroll
    for (int r = 0; r < 8; ++r) {
        float v = acc[r] + bn;           // logit for token (r + 8*half), expert e

        // top-1 over the 16-lane half-wave, lower index wins ties (jax top_k)
        float v1 = v; int i1 = e;
        #pragma unroll
        for (int m = 1; m < 16; m <<= 1) {
            float ov = __shfl_xor(v1, m, 16);
            int   oi = __shfl_xor(i1, m, 16);
            if (ov > v1 || (ov == v1 && oi < i1)) { v1 = ov; i1 = oi; }
        }
        // top-2: mask out winner, reduce again
        float v2 = (e == i1) ? -3.0e38f : v;
        int   i2 = e;
        #pragma unroll
        for (int m = 1; m < 16; m <<= 1) {
            float ov = __shfl_xor(v2, m, 16);
            int   oi = __shfl_xor(i2, m, 16);
            if (ov > v2 || (ov == v2 && oi < i2)) { v2 = ov; i2 = oi; }
        }

        // logsumexp over the two kept logits; others are -inf in sparse softmax
        float lse = v1 + log1pf(__expf(v2 - v1));
        int   t   = tile * 16 + r + half * 8;

        float p = (e == i1 || e == i2) ? __expf(v - lse) : 0.0f;
        rout[(size_t)t * E_DIM + e] = p;
        if (e == 0) {
            idx_out[(size_t)t * 2 + 0] = (float)i1;
            idx_out[(size_t)t * 2 + 1] = (float)i2;
        }
        lse2_sum += (e == 0) ? lse * lse : 0.0f;
    }

    // combine the two half-wave (lane 0 / lane 16) accumulators; lane 0 writes
    float tot = lse2_sum + __shfl_xor(lse2_sum, 16, 32);
    if (lane == 0) partials[tile] = tot;
}

// ---------------------------------------------------------------------------
// Kernel 3: deterministic reduction of per-tile lse^2 sums -> z_loss
// ---------------------------------------------------------------------------
__global__ __launch_bounds__(256) void zloss_kernel(
    const float* __restrict__ partials, float* __restrict__ zout) {
    __shared__ float sm[256];
    int t = threadIdx.x;
    float s = 0.0f;
    for (int i = t; i < NTILE; i += 256) s += partials[i];
    sm[t] = s;
    __syncthreads();
    for (int o = 128; o > 0; o >>= 1) {
        if (t < o) sm[t] += sm[t + o];
        __syncthreads();
    }
    if (t == 0) zout[0] = sm[0] / (float)NTOK;
}

// ---------------------------------------------------------------------------
extern "C" void kernel_launch(void* const* d_in, const int* in_sizes, int n_in,
                              void* d_out, int out_size, void* d_ws, size_t ws_size,
                              hipStream_t stream) {
    const float* mh      = (const float*)d_in[0];  // [4,4096,2048]
    const float* weight  = (const float*)d_in[1];  // [16,2048]
    const float* sigw    = (const float*)d_in[2];  // [16,2048]
    const float* bias    = (const float*)d_in[3];  // [16]
    const float* sigb    = (const float*)d_in[4];  // [16]
    const float* eps_in  = (const float*)d_in[5];  // [2048]
    const float* eps_out = (const float*)d_in[6];  // [16]

    float* out  = (float*)d_out;
    float* rout = out;                        // 262144 floats
    float* idxo = out + ROUT_CNT;             // 32768 floats
    float* zo   = out + ROUT_CNT + IDX_CNT;   // 1 float

    float* ws   = (float*)d_ws;
    float* wn   = ws;                         // 32768 floats
    float* bn   = ws + E_DIM * D_DIM;         // 16 floats
    float* part = bn + E_DIM;                 // 1024 floats

    prep_kernel<<<(E_DIM * D_DIM + 255) / 256, 256, 0, stream>>>(
        weight, sigw, bias, sigb, eps_in, eps_out, wn, bn);

    router_kernel<<<NTILE, TPB, 0, stream>>>(mh, wn, bn, rout, idxo, part);

    zloss_kernel<<<1, 256, 0, stream>>>(part, zo);
}